// RATGNN_26663156973810
// MI455X (gfx1250) — compile-verified
//
#include <hip/hip_runtime.h>
#include <math.h>

typedef __attribute__((ext_vector_type(2))) float v2f;
typedef __attribute__((ext_vector_type(8))) float v8f;

#define WMMA_F32X4(a, b, c) \
  __builtin_amdgcn_wmma_f32_16x16x4_f32(false, (a), false, (b), (short)0, (c), false, false)

// ---------------- constants (match reference) ----------------
#define NN  16384
#define FF  1024
#define LL  64
#define HH  256
#define SS  4096

// workspace layout (float offsets)
#define OFF_SUBEMB  0          // 64
#define OFF_TMP     64         // 256
#define OFF_TARFEAT 320        // 64
#define OFF_GE      384        // 2240
#define OFF_INJ     2624       // 1024
#define OFF_TARXW   3648       // 64
#define OFF_ADDXW   3712       // 64
#define OFF_BASE    3776       // 512
#define OFF_W12     4288       // 1024*64
#define OFF_SUBXW   69824      // 4096*64
#define OFF_LOGITS  331968     // 4096
#define OFF_TOPK    336064     // 3*4096

// ---------------- small helper kernels ----------------

// per-column mean of gathered node_emb rows: out[c] = mean_s node_emb[idx[s]][c]
__global__ void k_colmean(const float* __restrict__ node_emb,
                          const int* __restrict__ idx,
                          float* __restrict__ out) {
  const int c = blockIdx.x;             // 0..63
  float s = 0.f;
  for (int i = threadIdx.x; i < SS; i += blockDim.x)
    s += node_emb[(long)idx[i] * LL + c];
  __shared__ float red[256];
  red[threadIdx.x] = s;
  __syncthreads();
  for (int off = 128; off > 0; off >>= 1) {
    if (threadIdx.x < off) red[threadIdx.x] += red[threadIdx.x + off];
    __syncthreads();
  }
  if (threadIdx.x == 0) out[c] = red[0] * (1.0f / (float)SS);
}

// W12 = weight1 @ weight2   [1024,256]@[256,64] -> [1024,64]
__global__ void k_w12(const float* __restrict__ W1, const float* __restrict__ W2,
                      float* __restrict__ W12) {
  const int t = blockIdx.x * blockDim.x + threadIdx.x;  // 1024*64
  const int k = t >> 6, n = t & 63;
  float s = 0.f;
  for (int h = 0; h < HH; ++h) s += W1[k * HH + h] * W2[h * LL + n];
  W12[t] = s;
}

// generic matvec: y[j] = act( sum_k x[k]*W[k*J+j] + b[j] ); optional y2 = sigmoid(pre-act)
// act: 0=none, 1=relu, 2=leaky(0.01)
__global__ void k_matvec(const float* __restrict__ x, const int* __restrict__ rowsel,
                         int rowlen, const float* __restrict__ W,
                         const float* __restrict__ b, float* __restrict__ y,
                         float* __restrict__ y2, int K, int J, int act) {
  const int j = blockIdx.x * blockDim.x + threadIdx.x;
  if (j >= J) return;
  const float* xp = rowsel ? (x + (long)rowsel[0] * rowlen) : x;
  float s = b ? b[j] : 0.f;
  for (int k = 0; k < K; ++k) s += xp[k] * W[(long)k * J + j];
  float v = s;
  if (act == 1) v = fmaxf(v, 0.f);
  else if (act == 2) v = (v >= 0.f) ? v : 0.01f * v;
  y[j] = v;
  if (y2) y2[j] = 1.f / (1.f + expf(-s));
}

// graph_emb = [sub_emb(64), node_emb[target](64), tarfeat(64), wlabel(1024), wsec(1024)]
__global__ void k_assemble(const float* __restrict__ sub_emb,
                           const float* __restrict__ node_emb,
                           const int* __restrict__ tsel,
                           const float* __restrict__ tarfeat,
                           const float* __restrict__ wlabel,
                           const float* __restrict__ wsec,
                           float* __restrict__ ge) {
  const int i = blockIdx.x * blockDim.x + threadIdx.x;
  if (i >= 2240) return;
  float v;
  if (i < 64)        v = sub_emb[i];
  else if (i < 128)  v = node_emb[(long)tsel[0] * LL + (i - 64)];
  else if (i < 192)  v = tarfeat[i - 128];
  else if (i < 1216) v = wlabel[i - 192];
  else               v = wsec[i - 1216];
  ge[i] = v;
}

// base[j] = e_b1[j] + tar_xw . eW1[0:64,j] + add_xw . eW1[128:192,j]
//         + wlabel . eW1[193:1217,j] + wsec . eW1[1217:2241,j]
// (cat column layout: tar 0:64, sub 64:128, add 128:192, norm_a 192, wlabel 193:1217, wsec 1217:2241)
__global__ void k_base(const float* __restrict__ tar_xw, const float* __restrict__ add_xw,
                       const float* __restrict__ wlabel, const float* __restrict__ wsec,
                       const float* __restrict__ eW1, const float* __restrict__ eb1,
                       float* __restrict__ base) {
  const int j = blockIdx.x * blockDim.x + threadIdx.x;
  if (j >= 512) return;
  float s = eb1[j];
  for (int k = 0; k < 64; ++k)   s += tar_xw[k] * eW1[k * 512 + j];
  for (int k = 0; k < 64; ++k)   s += add_xw[k] * eW1[(128 + k) * 512 + j];
  for (int f = 0; f < 1024; ++f) s += wlabel[f] * eW1[(193 + f) * 512 + j];
  for (int f = 0; f < 1024; ++f) s += wsec[f]   * eW1[(1217 + f) * 512 + j];
  base[j] = s;
}

// ---------------- WMMA kernels (fp32 16x16x4) ----------------

// sub_xw = feat[idx] @ W12 : [4096,1024]@[1024,64] -> [4096,64]
// grid: (S/16 Mtiles, 64/16 Ntiles), 32 threads (one wave) per block.
__global__ void __launch_bounds__(32)
g1_gather_gemm(const float* __restrict__ feat, const int* __restrict__ idx,
               const float* __restrict__ W12, float* __restrict__ out) {
  const int mt = blockIdx.x, nt = blockIdx.y;
  const int lane = threadIdx.x;
  const int m  = lane & 15;
  const int kh = lane >> 4;             // 0: K={0,1}, 1: K={2,3}
  const int n  = nt * 16 + m;
  const float* arow = feat + (long)idx[mt * 16 + m] * FF;
  v8f acc = {};
#pragma unroll 4
  for (int k0 = 0; k0 < FF; k0 += 4) {
    const int ka = k0 + 2 * kh;
    v2f a; a.x = arow[ka];               a.y = arow[ka + 1];
    v2f b; b.x = W12[ka * LL + n];       b.y = W12[(ka + 1) * LL + n];
    acc = WMMA_F32X4(a, b, acc);
  }
  const int mrow = mt * 16 + ((lane < 16) ? 0 : 8);
#pragma unroll
  for (int i = 0; i < 8; ++i)
    out[(long)(mrow + i) * LL + (nt * 16 + m)] = acc[i];
}

// fused edge MLP per 16-row tile:
//  h1 = leaky( sub_xw_tile @ eW1[64:128,:] + base + norm_a * eW1[192,:] )   [16,512]
//  h2 = leaky( h1 @ eW2 + eb2 )                                             [16,32]
//  logit = h2 @ eW3 + eb3                                                   [16]
__global__ void __launch_bounds__(128)
g2_edge_mlp(const float* __restrict__ sub_xw, const float* __restrict__ eW1,
            const float* __restrict__ base, const float* __restrict__ eW2,
            const float* __restrict__ eb2, const float* __restrict__ eW3,
            const float* __restrict__ eb3, const float* __restrict__ adj,
            const int* __restrict__ idx, const int* __restrict__ tsel,
            float* __restrict__ logits) {
  __shared__ float A[16 * 64];
  __shared__ float h1[16 * 512];
  __shared__ float h2[16 * 32];
  __shared__ float na[16];
  const int tid  = threadIdx.x;
  const int lane = tid & 31;
  const int wave = tid >> 5;
  const int mt   = blockIdx.x;

  if (tid < 16)
    na[tid] = adj[(long)idx[mt * 16 + tid] * NN + tsel[0]];
  for (int t = tid; t < 16 * 64; t += 128) {
    const int r = t >> 6, c = t & 63;
    A[t] = sub_xw[(long)(mt * 16 + r) * LL + c];
  }
  __syncthreads();

  const int m  = lane & 15;
  const int kh = lane >> 4;
  const int mb = (lane < 16) ? 0 : 8;
  const float* w_na = eW1 + 192 * 512;

  // layer 1: 32 N-tiles split across 4 waves, K=64
  for (int nt = wave; nt < 32; nt += 4) {
    const int n = nt * 16 + m;
    v8f acc = {};
#pragma unroll
    for (int k0 = 0; k0 < 64; k0 += 4) {
      const int ka = k0 + 2 * kh;
      v2f a; a.x = A[m * 64 + ka];              a.y = A[m * 64 + ka + 1];
      v2f b; b.x = eW1[(64 + ka) * 512 + n];    b.y = eW1[(64 + ka + 1) * 512 + n];
      acc = WMMA_F32X4(a, b, acc);
    }
#pragma unroll
    for (int i = 0; i < 8; ++i) {
      const int mm = mb + i;
      float v = acc[i] + base[n] + na[mm] * w_na[n];
      h1[mm * 512 + n] = (v >= 0.f) ? v : 0.01f * v;
    }
  }
  __syncthreads();

  // layer 2: N=32 (2 tiles on waves 0,1), K=512 from LDS
  if (wave < 2) {
    const int n = wave * 16 + m;
    v8f acc = {};
#pragma unroll 8
    for (int k0 = 0; k0 < 512; k0 += 4) {
      const int ka = k0 + 2 * kh;
      v2f a; a.x = h1[m * 512 + ka];       a.y = h1[m * 512 + ka + 1];
      v2f b; b.x = eW2[ka * 32 + n];       b.y = eW2[(ka + 1) * 32 + n];
      acc = WMMA_F32X4(a, b, acc);
    }
#pragma unroll
    for (int i = 0; i < 8; ++i) {
      const int mm = mb + i;
      float v = acc[i] + eb2[n];
      h2[mm * 32 + n] = (v >= 0.f) ? v : 0.01f * v;
    }
  }
  __syncthreads();

  if (tid < 16) {
    float s = eb3[0];
#pragma unroll
    for (int k = 0; k < 32; ++k) s += h2[tid * 32 + k] * eW3[k];
    logits[mt * 16 + tid] = s;
  }
}

// ---------------- softmax + iterative top-k + outputs ----------------
// single block, 1024 threads. work: [0:S)=soft, [S:2S)=mutable copy, [2S:3S)=hard
__global__ void __launch_bounds__(1024)
k_softmax_topk(const float* __restrict__ logits, const int* __restrict__ subnodes,
               const int* __restrict__ budget, float* __restrict__ out,
               float* __restrict__ work) {
  const int tid = threadIdx.x;
  const int T = blockDim.x;
  __shared__ float red[1024];
  __shared__ int redi[1024];

  float mx = -INFINITY;
  for (int i = tid; i < SS; i += T) mx = fmaxf(mx, logits[i]);
  red[tid] = mx; __syncthreads();
  for (int off = T / 2; off > 0; off >>= 1) {
    if (tid < off) red[tid] = fmaxf(red[tid], red[tid + off]);
    __syncthreads();
  }
  mx = red[0]; __syncthreads();

  float sum = 0.f;
  for (int i = tid; i < SS; i += T) {
    const float e = expf(logits[i] - mx);
    work[i] = e; sum += e;
  }
  red[tid] = sum; __syncthreads();
  for (int off = T / 2; off > 0; off >>= 1) {
    if (tid < off) red[tid] += red[tid + off];
    __syncthreads();
  }
  const float inv = 1.f / red[0]; __syncthreads();

  for (int i = tid; i < SS; i += T) {
    const float s = work[i] * inv;
    work[i] = s; work[SS + i] = s; work[2 * SS + i] = 0.f;
  }
  __syncthreads();

  const int B = budget[0];
  for (int iter = 0; iter < B; ++iter) {
    float bv = -INFINITY; int bi = SS;
    for (int i = tid; i < SS; i += T) {
      const float v = work[SS + i];
      if (v > bv || (v == bv && i < bi)) { bv = v; bi = i; }
    }
    red[tid] = bv; redi[tid] = bi; __syncthreads();
    for (int off = T / 2; off > 0; off >>= 1) {
      if (tid < off) {
        const float ov = red[tid + off]; const int oi = redi[tid + off];
        if (ov > red[tid] || (ov == red[tid] && oi < redi[tid])) {
          red[tid] = ov; redi[tid] = oi;
        }
      }
      __syncthreads();
    }
    if (tid == 0) { work[2 * SS + redi[0]] = 1.f; work[SS + redi[0]] = -INFINITY; }
    __syncthreads();
  }

  for (int i = tid; i < SS; i += T) {
    const float soft = work[i];
    out[FF + i] = (work[2 * SS + i] - soft) + soft;       // score
    out[FF + SS + i] = (float)subnodes[i];                // masked_score_idx
  }
}

// ---------------- launch ----------------
extern "C" void kernel_launch(void* const* d_in, const int* in_sizes, int n_in,
                              void* d_out, int out_size, void* d_ws, size_t ws_size,
                              hipStream_t stream) {
  (void)in_sizes; (void)n_in; (void)out_size; (void)ws_size;
  const int*   target  = (const int*)d_in[0];
  const int*   nodes   = (const int*)d_in[1];
  const int*   budget  = (const int*)d_in[2];
  const float* feat    = (const float*)d_in[3];
  const float* adj     = (const float*)d_in[4];
  const float* nodeemb = (const float*)d_in[5];
  const float* wlabel  = (const float*)d_in[6];
  const float* wsec    = (const float*)d_in[7];
  const float* W1      = (const float*)d_in[8];
  const float* W2      = (const float*)d_in[9];
  const float* aW1     = (const float*)d_in[10];
  const float* ab1     = (const float*)d_in[11];
  const float* aW2     = (const float*)d_in[12];
  const float* ab2     = (const float*)d_in[13];
  const float* aW3     = (const float*)d_in[14];
  const float* ab3     = (const float*)d_in[15];
  const float* eW1     = (const float*)d_in[16];
  const float* eb1     = (const float*)d_in[17];
  const float* eW2     = (const float*)d_in[18];
  const float* eb2     = (const float*)d_in[19];
  const float* eW3     = (const float*)d_in[20];
  const float* eb3     = (const float*)d_in[21];

  float* ws  = (float*)d_ws;
  float* outp = (float*)d_out;
  float* sub_emb = ws + OFF_SUBEMB;
  float* tmp     = ws + OFF_TMP;
  float* tarfeat = ws + OFF_TARFEAT;
  float* ge      = ws + OFF_GE;
  float* inj     = ws + OFF_INJ;
  float* tar_xw  = ws + OFF_TARXW;
  float* add_xw  = ws + OFF_ADDXW;
  float* base    = ws + OFF_BASE;
  float* W12     = ws + OFF_W12;
  float* sub_xw  = ws + OFF_SUBXW;
  float* logits  = ws + OFF_LOGITS;
  float* topkwrk = ws + OFF_TOPK;

  // independent precomputes
  k_colmean<<<LL, 256, 0, stream>>>(nodeemb, nodes, sub_emb);
  k_w12<<<(FF * LL) / 256, 256, 0, stream>>>(W1, W2, W12);

  // AttrGeneration
  k_matvec<<<1, 256, 0, stream>>>(feat, target, FF, W1, nullptr, tmp, nullptr, FF, HH, 1);
  k_matvec<<<1, 64, 0, stream>>>(tmp, nullptr, 0, W2, nullptr, tarfeat, nullptr, HH, LL, 0);
  k_assemble<<<(2240 + 255) / 256, 256, 0, stream>>>(sub_emb, nodeemb, target, tarfeat,
                                                     wlabel, wsec, ge);
  // a-MLP (h_a1 reuses tmp region? no: keep tmp; reuse topk scratch)
  float* h_a1 = topkwrk;            // transient reuse of topk scratch
  float* h_a2 = topkwrk + 512;
  k_matvec<<<1, 128, 0, stream>>>(ge, nullptr, 0, aW1, ab1, h_a1, nullptr, 2240, 128, 2);
  k_matvec<<<2, 256, 0, stream>>>(h_a1, nullptr, 0, aW2, ab2, h_a2, nullptr, 128, 512, 2);
  k_matvec<<<4, 256, 0, stream>>>(h_a2, nullptr, 0, aW3, ab3, outp /*add_feat*/, inj,
                                  512, FF, 0);

  // xw vectors (no relu path uses fused W12)
  k_matvec<<<1, 64, 0, stream>>>(feat, target, FF, W12, nullptr, tar_xw, nullptr, FF, LL, 0);
  k_matvec<<<1, 64, 0, stream>>>(inj, nullptr, 0, W12, nullptr, add_xw, nullptr, FF, LL, 0);
  k_base<<<2, 256, 0, stream>>>(tar_xw, add_xw, wlabel, wsec, eW1, eb1, base);

  // big fused path
  dim3 g1grid(SS / 16, LL / 16);
  g1_gather_gemm<<<g1grid, 32, 0, stream>>>(feat, nodes, W12, sub_xw);
  g2_edge_mlp<<<SS / 16, 128, 0, stream>>>(sub_xw, eW1, base, eW2, eb2, eW3, eb3,
                                           adj, nodes, target, logits);

  // softmax + straight-through top-k + outputs
  k_softmax_topk<<<1, 1024, 0, stream>>>(logits, nodes, budget, outp, topkwrk);
}